// TripletLoss_77403900608666
// MI455X (gfx1250) — compile-verified
//
#include <hip/hip_runtime.h>
#include <hip/hip_bf16.h>

#define NN   512
#define DD   256
#define MARG 0.2f

typedef __attribute__((ext_vector_type(2))) float v2f;
typedef __attribute__((ext_vector_type(8))) float v8f;

// ---------------------------------------------------------------------------
// Threefry-2x32 (exact JAX schedule): ks2 = k0 ^ k1 ^ 0x1BD11BDA, 20 rounds.
// ---------------------------------------------------------------------------
__device__ __forceinline__ unsigned rotl32(unsigned x, int r) {
    return (x << r) | (x >> (32 - r));
}

__device__ __forceinline__ void threefry2x32(unsigned k0, unsigned k1,
                                             unsigned c0, unsigned c1,
                                             unsigned& o0, unsigned& o1) {
    unsigned ks0 = k0, ks1 = k1, ks2 = k0 ^ k1 ^ 0x1BD11BDAu;
    unsigned x0 = c0 + ks0, x1 = c1 + ks1;
#define TF_ROUND(r) { x0 += x1; x1 = rotl32(x1, (r)); x1 ^= x0; }
    TF_ROUND(13) TF_ROUND(15) TF_ROUND(26) TF_ROUND(6)
    x0 += ks1; x1 += ks2 + 1u;
    TF_ROUND(17) TF_ROUND(29) TF_ROUND(16) TF_ROUND(24)
    x0 += ks2; x1 += ks0 + 2u;
    TF_ROUND(13) TF_ROUND(15) TF_ROUND(26) TF_ROUND(6)
    x0 += ks0; x1 += ks1 + 3u;
    TF_ROUND(17) TF_ROUND(29) TF_ROUND(16) TF_ROUND(24)
    x0 += ks1; x1 += ks2 + 4u;
    TF_ROUND(13) TF_ROUND(15) TF_ROUND(26) TF_ROUND(6)
    x0 += ks2; x1 += ks0 + 5u;
#undef TF_ROUND
    o0 = x0; o1 = x1;
}

// Gumbel noise for flat index idx in [0, 512^3), key = (0, 42).
// JAX counter-mode split-halves: out[i] = x0 of (i, i+half); out[i+half] = x1.
__device__ __forceinline__ float gumbel_at(unsigned idx) {
    const unsigned HALF = 67108864u;  // 512^3 / 2
    unsigned c0, c1, o0, o1;
    bool second = (idx >= HALF);
    c0 = second ? (idx - HALF) : idx;
    c1 = second ? idx : (idx + HALF);
    threefry2x32(0u, 42u, c0, c1, o0, o1);
    unsigned bits = second ? o1 : o0;
    float u = __uint_as_float((bits >> 9) | 0x3F800000u) - 1.0f;   // [0,1)
    const float TINY = 1.17549435e-38f;
    u = u * (1.0f - TINY) + TINY;          // rescale to [tiny, 1)
    u = fmaxf(u, TINY);
    return -logf(-logf(u));
}

// ---------------------------------------------------------------------------
// Kernel 1: G = E * E^T via V_WMMA_F32_16X16X4_F32. One wave -> one 16x16 tile.
// A(16x4): lane<16 holds row=lane, K={0,1}; lane>=16 holds row=lane-16, K={2,3}.
// B(4x16) = E^T slice: same float2 pattern with the column-tile row base.
// ---------------------------------------------------------------------------
__global__ __launch_bounds__(256) void gram_wmma_kernel(
        const float* __restrict__ E, float* __restrict__ G) {
    const int lane = threadIdx.x & 31;
    const int wave = threadIdx.x >> 5;
    const int tile = blockIdx.x * 8 + wave;      // 0..1023
    const int tr = tile >> 5;                    // tile row (16-row units)
    const int tc = tile & 31;                    // tile col
    const int l16 = lane & 15;
    const int hi  = lane >> 4;                   // 0: K 0,1   1: K 2,3

    const float* arow = E + (tr * 16 + l16) * DD + hi * 2;
    const float* brow = E + (tc * 16 + l16) * DD + hi * 2;

    v8f acc = {};
#pragma unroll 4
    for (int k = 0; k < DD; k += 4) {
        v2f a = *(const v2f*)(arow + k);
        v2f b = *(const v2f*)(brow + k);
        acc = __builtin_amdgcn_wmma_f32_16x16x4_f32(
            /*neg_a=*/false, a, /*neg_b=*/false, b,
            /*c_mod=*/(short)0, acc, /*reuse_a=*/false, /*reuse_b=*/false);
    }

    // C/D layout: VGPR v -> (M = v + 8*hi, N = l16)
    const int r0 = tr * 16 + hi * 8;
    const int c0 = tc * 16 + l16;
#pragma unroll
    for (int v = 0; v < 8; ++v)
        G[(r0 + v) * NN + c0] = acc[v];
}

// ---------------------------------------------------------------------------
// Kernel 2: sqd[i,j] = max(G[i,i] + G[j,j] - 2 G[i,j], 0)
// ---------------------------------------------------------------------------
__global__ __launch_bounds__(256) void sqd_kernel(
        const float* __restrict__ G, float* __restrict__ sqd) {
    int idx = blockIdx.x * 256 + threadIdx.x;    // 0 .. 512*512-1
    int i = idx >> 9, j = idx & (NN - 1);
    float v = G[i * NN + i] + G[j * NN + j] - 2.0f * G[idx];
    sqd[idx] = fmaxf(v, 0.0f);
}

__global__ void zero_kernel(float* __restrict__ p) {
    if (threadIdx.x < 2) p[threadIdx.x] = 0.0f;
}

// ---------------------------------------------------------------------------
// Kernel 3: one wave per (a,p). Semi-hard window scan over j, Gumbel argmax
// (threefry only evaluated where valid), wave argmax reduce (ties -> lower j),
// then atomic accumulate of term and count.
// ---------------------------------------------------------------------------
__global__ __launch_bounds__(256) void triplet_kernel(
        const float* __restrict__ sqd, const int* __restrict__ labels,
        float* __restrict__ sums) {
    const int a    = blockIdx.x;
    const int p    = blockIdx.y * 8 + (threadIdx.x >> 5);
    const int lane = threadIdx.x & 31;
    if (a == p) return;                      // wave-uniform
    const int la = labels[a];
    if (labels[p] != la) return;             // wave-uniform

    const float dap = sqd[a * NN + p];
    const float hiT = dap + MARG;
    const unsigned base = (unsigned)(a * NN + p) * NN;

    float best = -__builtin_inff();
    int   bj   = 0;

    for (int j = lane; j < NN; j += 32) {
        float daj = sqd[a * NN + j];
        if ((labels[j] != la) && (daj > dap) && (daj < hiT)) {
            float g = gumbel_at(base + (unsigned)j);
            if (g > best) { best = g; bj = j; }   // j ascending per lane
        }
    }
    // wave32 argmax reduction; tie-break toward smaller j (argmax semantics)
    for (int off = 16; off > 0; off >>= 1) {
        float ob = __shfl_xor(best, off, 32);
        int   oj = __shfl_xor(bj,   off, 32);
        if (ob > best || (ob == best && oj < bj)) { best = ob; bj = oj; }
    }
    if (lane == 0 && best > -__builtin_inff()) {
        atomicAdd(&sums[0], dap - sqd[a * NN + bj] + MARG);
        atomicAdd(&sums[1], 1.0f);
    }
}

__global__ void finalize_kernel(const float* __restrict__ sums,
                                float* __restrict__ out) {
    out[0] = sums[0] / fmaxf(sums[1], 1.0f);
}

// ---------------------------------------------------------------------------
extern "C" void kernel_launch(void* const* d_in, const int* in_sizes, int n_in,
                              void* d_out, int out_size, void* d_ws, size_t ws_size,
                              hipStream_t stream) {
    const float* E      = (const float*)d_in[0];   // (512, 256) fp32
    const int*   labels = (const int*)d_in[1];     // (512,) int
    float* out = (float*)d_out;

    float* ws   = (float*)d_ws;
    float* G    = ws;                 // 512*512 floats
    float* sqd  = ws + NN * NN;       // 512*512 floats
    float* sums = ws + 2 * NN * NN;   // [sum, count]

    zero_kernel<<<1, 32, 0, stream>>>(sums);
    gram_wmma_kernel<<<128, 256, 0, stream>>>(E, G);          // 1024 tiles
    sqd_kernel<<<(NN * NN) / 256, 256, 0, stream>>>(G, sqd);
    triplet_kernel<<<dim3(NN, NN / 8), 256, 0, stream>>>(sqd, labels, sums);
    finalize_kernel<<<1, 1, 0, stream>>>(sums, out);
    (void)in_sizes; (void)n_in; (void)out_size; (void)ws_size;
}